// HMMEstimator_62912680951978
// MI455X (gfx1250) — compile-verified
//
#include <hip/hip_runtime.h>

typedef float v2f __attribute__((ext_vector_type(2)));
typedef float v8f __attribute__((ext_vector_type(8)));

#define HW    65536
#define SLICE 196608   // HW * 3

__device__ __forceinline__ float lane_bcast(float v, int srcLane) {
  // wave32 lane gather via LDS permute unit (no LDS memory touched)
  return __int_as_float(__builtin_amdgcn_ds_bpermute(srcLane << 2, __float_as_int(v)));
}
__device__ __forceinline__ float nt_load(const float* p) { return __builtin_nontemporal_load(p); }
__device__ __forceinline__ void  nt_store(float* p, float v) { __builtin_nontemporal_store(v, p); }

__global__ __launch_bounds__(256) void hmm_step_kernel(
    const float* __restrict__ obs,       // [HW][3]
    const float* __restrict__ param_obs, // [3][3]
    const float* __restrict__ tw,        // [3][3][3][3] OIHW
    const float* __restrict__ tb,        // [3]
    const float* __restrict__ u_in,      // [HW][3]
    const float* __restrict__ wO_in,     // [9][HW][3]
    const float* __restrict__ wT_in,     // [81][HW][3]
    const float* __restrict__ wB_in,     // [3][HW][3]
    float* __restrict__ out)
{
  const int tid  = threadIdx.x;
  const int p    = blockIdx.x * 256 + tid;
  const int h    = p >> 8;
  const int w    = p & 255;
  const int lane = tid & 31;
  const int wave = tid >> 5;
  const int half = lane >> 4;
  const int lsub = lane & 15;

  // ---- S = softmax(param_obs, axis=1); uniform 3x3 (scalar-cached loads) ----
  float S[3][3];
  #pragma unroll
  for (int i = 0; i < 3; ++i) {
    float l0 = param_obs[i*3+0], l1 = param_obs[i*3+1], l2 = param_obs[i*3+2];
    float m  = fmaxf(l0, fmaxf(l1, l2));
    float e0 = __expf(l0 - m), e1 = __expf(l1 - m), e2 = __expf(l2 - m);
    float inv = 1.0f / (e0 + e1 + e2);
    S[i][0] = e0*inv; S[i][1] = e1*inv; S[i][2] = e2*inv;
  }

  // ---- per-pixel observation ----
  float ob[3];
  ob[0] = obs[p*3+0]; ob[1] = obs[p*3+1]; ob[2] = obs[p*3+2];

  // ---- b = S * obs via V_WMMA_F32_16X16X4_F32 (two 16-pixel tiles per wave) ----
  // A (16x4): M = wave-local pixel, K = obs channel (K=3 padded 0).
  //   lanes 0-15: vgpr0=K0, vgpr1=K1 ; lanes 16-31: vgpr0=K2, vgpr1=K3(=0)
  // B (4x16): B[K][N] = S[N][K]  (N = state), rows K0/K2 in vgpr0 halves, K1/K3 in vgpr1.
  // D (16x16): b[pixel M][state N] -> scatter via LDS so each lane reads its own pixel.
  __shared__ float bshr[8][32][3];
  v2f bf;
  {
    float b0 = 0.f, b1 = 0.f;
    if (lsub < 3) {
      if (half == 0) { b0 = S[lsub][0]; b1 = S[lsub][1]; }
      else           { b0 = S[lsub][2]; b1 = 0.f;        }
    }
    bf[0] = b0; bf[1] = b1;
  }
  #pragma unroll
  for (int g = 0; g < 2; ++g) {
    const int src = g*16 + lsub;
    float so0 = lane_bcast(ob[0], src);
    float so1 = lane_bcast(ob[1], src);
    float so2 = lane_bcast(ob[2], src);
    v2f af;
    af[0] = (half == 0) ? so0 : so2;
    af[1] = (half == 0) ? so1 : 0.f;
    v8f c = {0.f,0.f,0.f,0.f,0.f,0.f,0.f,0.f};
    v8f d = __builtin_amdgcn_wmma_f32_16x16x4_f32(false, af, false, bf,
                                                  (short)0, c, false, false);
    if (lsub < 3) {
      #pragma unroll
      for (int r = 0; r < 8; ++r)
        bshr[wave][g*16 + half*8 + r][lsub] = d[r];   // D: vgpr r -> M = r + 8*half
    }
  }
  __syncthreads();

  float b[3];
  b[0] = bshr[wave][lane][0]; b[1] = bshr[wave][lane][1]; b[2] = bshr[wave][lane][2];

  // ---- filtering update ----
  float u0 = u_in[p*3+0], u1 = u_in[p*3+1], u2 = u_in[p*3+2];
  float uarr[3] = {u0, u1, u2};
  float Bu[3]   = {b[0]*u0, b[1]*u1, b[2]*u2};
  float bu      = Bu[0] + Bu[1] + Bu[2];
  float inv_bu  = 1.0f / bu;
  float q[3]    = {Bu[0]*inv_bu, Bu[1]*inv_bu, Bu[2]*inv_bu};

  // ---- boundary tap masks; Wsum; P = softmax_i(Wsum + bias) ----
  float vh[3] = { (h > 0) ? 1.f : 0.f, 1.f, (h < 255) ? 1.f : 0.f };
  float vw[3] = { (w > 0) ? 1.f : 0.f, 1.f, (w < 255) ? 1.f : 0.f };

  float P[3][3];
  {
    float L[3][3];
    #pragma unroll
    for (int i = 0; i < 3; ++i)
      #pragma unroll
      for (int j = 0; j < 3; ++j) {
        float acc = tb[i];
        #pragma unroll
        for (int kh = 0; kh < 3; ++kh)
          #pragma unroll
          for (int kw = 0; kw < 3; ++kw)
            acc = fmaf(vh[kh]*vw[kw], tw[((i*3+j)*3+kh)*3+kw], acc);
        L[i][j] = acc;
      }
    #pragma unroll
    for (int j = 0; j < 3; ++j) {
      float m  = fmaxf(L[0][j], fmaxf(L[1][j], L[2][j]));
      float e0 = __expf(L[0][j]-m), e1 = __expf(L[1][j]-m), e2 = __expf(L[2][j]-m);
      float inv = 1.0f / (e0 + e1 + e2);
      P[0][j] = e0*inv; P[1][j] = e1*inv; P[2][j] = e2*inv;
    }
  }

  // ---- u_kp1, temp1, Phi (closed forms) ----
  float ukp1[3], T1[3][3], Phi[3][3];
  #pragma unroll
  for (int i = 0; i < 3; ++i)
    ukp1[i] = P[i][0]*q[0] + P[i][1]*q[1] + P[i][2]*q[2];
  #pragma unroll
  for (int i = 0; i < 3; ++i)
    #pragma unroll
    for (int j = 0; j < 3; ++j) {
      T1[i][j]  = P[i][j] - ukp1[i];
      Phi[i][j] = b[j] * T1[i][j] * inv_bu;
    }

  float* out_u = out;              // [HW][3]
  float* out_O = out + 1*SLICE;    // [9][HW][3]
  float* out_T = out + 10*SLICE;   // [81][HW][3]
  float* out_B = out + 91*SLICE;   // [3][HW][3]
  const size_t base = (size_t)p * 3;

  nt_store(out_u + base + 0, ukp1[0]);
  nt_store(out_u + base + 1, ukp1[1]);
  nt_store(out_u + base + 2, ukp1[2]);

  // ---- observation-parameter sensitivity: w' = Phi*w + T1[:,s]*(S[s][o](obs_o-b_s)u_s/bu) ----
  #pragma unroll
  for (int s = 0; s < 3; ++s)
    #pragma unroll
    for (int o = 0; o < 3; ++o) {
      const int sl = s*3 + o;
      const float* src = wO_in + (size_t)sl*SLICE + base;
      float*       dst = out_O + (size_t)sl*SLICE + base;
      float w0 = nt_load(src+0), w1 = nt_load(src+1), w2 = nt_load(src+2);
      float rs = S[s][o] * (ob[o] - b[s]) * uarr[s] * inv_bu;
      #pragma unroll
      for (int i = 0; i < 3; ++i) {
        float v = fmaf(Phi[i][0], w0, fmaf(Phi[i][1], w1, fmaf(Phi[i][2], w2, T1[i][s]*rs)));
        nt_store(dst+i, v);
      }
    }

  // ---- transition-weight sensitivity: w' = Phi*w + valid(kh,kw)*G[a][bb][i] ----
  #pragma unroll
  for (int a = 0; a < 3; ++a)
    #pragma unroll
    for (int bb = 0; bb < 3; ++bb) {
      float G[3];
      #pragma unroll
      for (int i = 0; i < 3; ++i)
        G[i] = q[bb] * P[i][bb] * (((i == a) ? 1.f : 0.f) - P[a][bb]);
      #pragma unroll
      for (int kh = 0; kh < 3; ++kh)
        #pragma unroll
        for (int kw = 0; kw < 3; ++kw) {
          const int sl = ((a*3 + bb)*3 + kh)*3 + kw;
          const float* src = wT_in + (size_t)sl*SLICE + base;
          float*       dst = out_T + (size_t)sl*SLICE + base;
          const float  vm  = vh[kh] * vw[kw];
          float w0 = nt_load(src+0), w1 = nt_load(src+1), w2 = nt_load(src+2);
          #pragma unroll
          for (int i = 0; i < 3; ++i) {
            float v = fmaf(Phi[i][0], w0, fmaf(Phi[i][1], w1, fmaf(Phi[i][2], w2, vm*G[i])));
            nt_store(dst+i, v);
          }
        }
    }

  // ---- transition-bias sensitivity: w' = Phi*w + (delta(i,a)ukp1_i - sum_j P_ij P_aj q_j) ----
  #pragma unroll
  for (int a = 0; a < 3; ++a) {
    const float* src = wB_in + (size_t)a*SLICE + base;
    float*       dst = out_B + (size_t)a*SLICE + base;
    float w0 = nt_load(src+0), w1 = nt_load(src+1), w2 = nt_load(src+2);
    #pragma unroll
    for (int i = 0; i < 3; ++i) {
      float Mi = P[i][0]*P[a][0]*q[0] + P[i][1]*P[a][1]*q[1] + P[i][2]*P[a][2]*q[2];
      float rb = ((i == a) ? ukp1[i] : 0.f) - Mi;
      float v  = fmaf(Phi[i][0], w0, fmaf(Phi[i][1], w1, fmaf(Phi[i][2], w2, rb)));
      nt_store(dst+i, v);
    }
  }
}

extern "C" void kernel_launch(void* const* d_in, const int* in_sizes, int n_in,
                              void* d_out, int out_size, void* d_ws, size_t ws_size,
                              hipStream_t stream) {
  (void)in_sizes; (void)n_in; (void)out_size; (void)d_ws; (void)ws_size;
  hmm_step_kernel<<<dim3(HW/256), dim3(256), 0, stream>>>(
      (const float*)d_in[0],  // obs
      (const float*)d_in[1],  // param_obs
      (const float*)d_in[2],  // trans_weight
      (const float*)d_in[3],  // trans_bias
      (const float*)d_in[4],  // u_k
      (const float*)d_in[5],  // w_k_O
      (const float*)d_in[6],  // w_k_T_weight
      (const float*)d_in[7],  // w_k_T_bias
      (float*)d_out);
}